// ConditionalSpline1DFlow_8383776162334
// MI455X (gfx1250) — compile-verified
//
#include <hip/hip_runtime.h>
#include <hip/hip_bf16.h>
#include <math.h>

// ---------------------------------------------------------------------------
// ConditionalSpline1DFlow for MI455X (gfx1250, wave32)
//   Phase 0: pad W3/b3 (49 -> 64 cols) into workspace (removes all guards
//            from the WMMA inner loops -> no EXEC-mask branching per load)
//   Phase 1: 3-layer MLP via V_WMMA_F32_16X16X4_F32 + spline param prep
//   Phase 2: memory-bound rational-quadratic spline apply (the real work)
// ---------------------------------------------------------------------------

typedef __attribute__((ext_vector_type(2))) float v2f;
typedef __attribute__((ext_vector_type(8))) float v8f;

#define BATCH   4096
#define NPTS    1024
#define CD      64
#define HID     256
#define NOUT    49          // 3*K + 1
#define NOUTP   64          // padded
#define KBINS   16
#define PSTRIDE 84          // per-row param stride (83 used)
#define BOUNDF  5.0f
#define MBWF    0.001f
#define MBHF    0.001f
#define MDF     0.001f

// workspace layout (floats)
#define WS_PARAMS 0
#define WS_W3PAD  ((size_t)BATCH * PSTRIDE)            // 344064
#define WS_B3PAD  (WS_W3PAD + (size_t)HID * NOUTP)     // + 16384

// ---- fp32 WMMA tile: D(16x16) += A(16x4) * B(4x16), K stepped by 4 --------
// A-frag (ISA 7.12.2, 32-bit A 16x4): lane L holds row M=L%16, k-pair 2*(L/16)
// B-frag (column-striped): lane L holds col N=n0+L%16, rows k..k+1 in .x/.y
template <int K>
__device__ inline v8f tile_gemm_f32(const float* actIn, int ldi,
                                    const float* __restrict__ W, int ldw,
                                    int n0, int lane) {
  const int g  = lane >> 4;      // 0 or 1
  const int nl = lane & 15;
  const int n  = n0 + nl;
  v8f acc = {};
#pragma unroll 4
  for (int s = 0; s < K / 4; ++s) {
    const int k = 4 * s + 2 * g;
    v2f a = *(const v2f*)(actIn + nl * ldi + k);      // 8B LDS load, aligned
    v2f b;
    b.x = W[(size_t)k       * ldw + n];
    b.y = W[(size_t)(k + 1) * ldw + n];
    // 8 args: (neg_a, A, neg_b, B, c_mod, C, reuse_a, reuse_b)
    acc = __builtin_amdgcn_wmma_f32_16x16x4_f32(false, a, false, b,
                                                (short)0, acc, false, false);
  }
  return acc;
}

// C/D layout: lane L holds col N=L%16; VGPR r holds row M = 8*(L/16) + r
__device__ inline void store_tile_relu(const v8f& acc, const float* bias,
                                       int n0, int lane,
                                       float* outBuf, int ldo, bool relu) {
  const int g  = lane >> 4;
  const int nl = lane & 15;
  const int n  = n0 + nl;
  const float bn = bias[n];
#pragma unroll
  for (int r = 0; r < 8; ++r) {
    float v = acc[r] + bn;
    if (relu) v = fmaxf(v, 0.0f);
    outBuf[(8 * g + r) * ldo + n] = v;
  }
}

__device__ inline float softplusf(float x) {
  return fmaxf(x, 0.0f) + log1pf(expf(-fabsf(x)));
}

// ---------------------------------------------------------------------------
// Kernel 0: zero-pad W3 (256x49 -> 256x64) and b3 (49 -> 64) into workspace.
// ---------------------------------------------------------------------------
__global__ void __launch_bounds__(256)
pad_w3_kernel(const float* __restrict__ W3, const float* __restrict__ b3,
              float* __restrict__ W3pad, float* __restrict__ b3pad) {
  const int e = blockIdx.x * 256 + threadIdx.x;       // 64 blocks -> 16384
  const int r = e >> 6, c = e & 63;
  W3pad[e] = (c < NOUT) ? W3[(size_t)r * NOUT + c] : 0.0f;
  if (blockIdx.x == 0 && threadIdx.x < NOUTP)
    b3pad[threadIdx.x] = (threadIdx.x < NOUT) ? b3[threadIdx.x] : 0.0f;
}

// ---------------------------------------------------------------------------
// Kernel 1: per-16-row M-tile, 8 waves split columns. MLP + spline params.
// ---------------------------------------------------------------------------
__global__ void __launch_bounds__(256)
mlp_params_kernel(const float* __restrict__ cond,
                  const float* __restrict__ W1, const float* __restrict__ b1,
                  const float* __restrict__ W2, const float* __restrict__ b2,
                  const float* __restrict__ W3pad,
                  const float* __restrict__ b3pad,
                  float* __restrict__ params) {
  __shared__ float sCond[16 * 66];    // 16 x 64, padded stride (bank spread)
  __shared__ float sA[16 * 258];      // 16 x 256 activations
  __shared__ float sB[16 * 258];
  __shared__ float sP[16 * 66];       // 16 x 64 logits (padded stride)

  const int tid  = threadIdx.x;
  const int lane = tid & 31;
  const int wave = tid >> 5;
  const int row0 = blockIdx.x * 16;

  // stage cond tile (16x64): 1024 floats, 4 per thread, coalesced
#pragma unroll
  for (int i = 0; i < 4; ++i) {
    int e = tid + 256 * i;
    int r = e >> 6, c = e & 63;
    sCond[r * 66 + c] = cond[(size_t)(row0 + r) * CD + c];
  }
  __syncthreads();

  // layer 1: 64 -> 256, relu. 16 column tiles over 8 waves.
#pragma unroll
  for (int t = 0; t < 2; ++t) {
    int n0 = (wave * 2 + t) * 16;
    v8f acc = tile_gemm_f32<CD>(sCond, 66, W1, HID, n0, lane);
    store_tile_relu(acc, b1, n0, lane, sA, 258, true);
  }
  __syncthreads();

  // layer 2: 256 -> 256, relu
#pragma unroll
  for (int t = 0; t < 2; ++t) {
    int n0 = (wave * 2 + t) * 16;
    v8f acc = tile_gemm_f32<HID>(sA, 258, W2, HID, n0, lane);
    store_tile_relu(acc, b2, n0, lane, sB, 258, true);
  }
  __syncthreads();

  // layer 3: 256 -> 64 (padded), no relu; waves 0..3 (uniform branch)
  if (wave < 4) {
    int n0 = wave * 16;
    v8f acc = tile_gemm_f32<HID>(sB, 258, W3pad, NOUTP, n0, lane);
    store_tile_relu(acc, b3pad, n0, lane, sP, 66, false);
  }
  __syncthreads();

  // spline parameter math: one thread per batch row of this tile
  if (tid < 16) {
    const float* p = &sP[tid * 66];
    float* pr = params + (size_t)(row0 + tid) * PSTRIDE;
    // layout: [cumw 0..16][cumh 17..33][deriv 34..50][w 51..66][h 67..82]

    // widths = MBW + (10 - 16*MBW) * softmax(p[0:16]); cumw = prefix + left
    {
      float m = -INFINITY;
#pragma unroll
      for (int i = 0; i < KBINS; ++i) m = fmaxf(m, p[i]);
      float e[KBINS], s = 0.0f;
#pragma unroll
      for (int i = 0; i < KBINS; ++i) { e[i] = expf(p[i] - m); s += e[i]; }
      const float scale = (2.0f * BOUNDF - KBINS * MBWF) / s;
      float cw = -BOUNDF;
      pr[0] = cw;
#pragma unroll
      for (int i = 0; i < KBINS; ++i) {
        float w = MBWF + scale * e[i];
        pr[51 + i] = w;
        cw += w;
        pr[1 + i] = cw;
      }
    }
    // heights / cumh from p[16:32]
    {
      float m = -INFINITY;
#pragma unroll
      for (int i = 0; i < KBINS; ++i) m = fmaxf(m, p[16 + i]);
      float e[KBINS], s = 0.0f;
#pragma unroll
      for (int i = 0; i < KBINS; ++i) { e[i] = expf(p[16 + i] - m); s += e[i]; }
      const float scale = (2.0f * BOUNDF - KBINS * MBHF) / s;
      float ch = -BOUNDF;
      pr[17] = ch;
#pragma unroll
      for (int i = 0; i < KBINS; ++i) {
        float h = MBHF + scale * e[i];
        pr[67 + i] = h;
        ch += h;
        pr[18 + i] = ch;
      }
    }
    // derivatives from p[32:49]
#pragma unroll
    for (int j = 0; j < KBINS + 1; ++j)
      pr[34 + j] = MDF + softplusf(p[32 + j]);
  }
}

// ---------------------------------------------------------------------------
// Kernel 2: one block per batch row; 256 threads x float4 = 1024 points.
// Pure streaming: 4 KB y in, 8 KB out per block; params broadcast via LDS.
// ---------------------------------------------------------------------------
__global__ void __launch_bounds__(256)
spline_apply_kernel(const float* __restrict__ y,
                    const float* __restrict__ params,
                    float* __restrict__ out,
                    float* __restrict__ logdet) {
  __shared__ float pp[PSTRIDE];
  const int tid = threadIdx.x;
  const int row = blockIdx.x;
  if (tid < 83) pp[tid] = params[(size_t)row * PSTRIDE + tid];
  __syncthreads();

  const float* cumw  = pp;
  const float* cumh  = pp + 17;
  const float* deriv = pp + 34;
  const float* wd    = pp + 51;
  const float* ht    = pp + 67;

  const size_t base = (size_t)row * NPTS + tid * 4;
  const float4 yv = *(const float4*)(y + base);
  const float yy[4] = {yv.x, yv.y, yv.z, yv.w};
  float o[4], ld[4];

#pragma unroll
  for (int i = 0; i < 4; ++i) {
    const float yi = yy[i];
    const float xc = fminf(fmaxf(yi, -BOUNDF), BOUNDF);

    int bin = 0;
#pragma unroll
    for (int j = 1; j <= KBINS; ++j) bin += (xc >= cumw[j]) ? 1 : 0;
    bin = (bin > KBINS - 1) ? (KBINS - 1) : bin;

    const float x0 = cumw[bin];
    const float w  = wd[bin];
    const float y0 = cumh[bin];
    const float hh = ht[bin];
    const float d0 = deriv[bin];
    const float d1 = deriv[bin + 1];

    const float delta = hh / w;
    const float th    = (xc - x0) / w;
    const float omt   = 1.0f - th;
    const float num   = hh * (delta * th * th + d0 * th * omt);
    const float den   = delta + (d0 + d1 - 2.0f * delta) * th * omt;
    const float outv  = y0 + num / den;

    const float dn = fmaxf(delta * delta *
                           (d1 * th * th + 2.0f * delta * th * omt + d0 * omt * omt),
                           1e-12f);
    const float dd = fmaxf(den * den, 1e-12f);
    const float l  = logf(dn) - logf(dd);

    const bool outside = (yi < -BOUNDF) || (yi > BOUNDF);
    o[i]  = outside ? yi : outv;
    ld[i] = outside ? 0.0f : l;
  }

  *(float4*)(out + base)    = make_float4(o[0], o[1], o[2], o[3]);
  *(float4*)(logdet + base) = make_float4(ld[0], ld[1], ld[2], ld[3]);
}

// ---------------------------------------------------------------------------
extern "C" void kernel_launch(void* const* d_in, const int* in_sizes, int n_in,
                              void* d_out, int out_size, void* d_ws, size_t ws_size,
                              hipStream_t stream) {
  const float* cond = (const float*)d_in[0];
  const float* yv   = (const float*)d_in[1];
  const float* W1   = (const float*)d_in[2];
  const float* b1   = (const float*)d_in[3];
  const float* W2   = (const float*)d_in[4];
  const float* b2   = (const float*)d_in[5];
  const float* W3   = (const float*)d_in[6];
  const float* b3   = (const float*)d_in[7];

  float* out    = (float*)d_out;                         // (B, N)
  float* logdet = out + (size_t)BATCH * NPTS;            // (B, N)

  float* wsf    = (float*)d_ws;
  float* params = wsf + WS_PARAMS;                       // B * PSTRIDE
  float* W3pad  = wsf + WS_W3PAD;                        // 256 * 64
  float* b3pad  = wsf + WS_B3PAD;                        // 64

  pad_w3_kernel<<<(HID * NOUTP) / 256, 256, 0, stream>>>(W3, b3, W3pad, b3pad);
  mlp_params_kernel<<<BATCH / 16, 256, 0, stream>>>(cond, W1, b1, W2, b2,
                                                    W3pad, b3pad, params);
  spline_apply_kernel<<<BATCH, 256, 0, stream>>>(yv, params, out, logdet);
}